// LegacyFacebookGNN_16853451669719
// MI455X (gfx1250) — compile-verified
//
#include <hip/hip_runtime.h>
#include <hip/hip_bf16.h>

// ---------------------------------------------------------------------------
// LegacyFacebookGNN on MI455X (gfx1250, wave32).
// Bottleneck: edge gather/scatter (~5 GB, L2-resident since N*128*4 = 51.2MB
// << 192MB L2). GEMMs (~23.5 GFLOP) done exactly in fp32 with
// v_wmma_f32_16x16x4_f32 since compute is nowhere near the roofline.
// ---------------------------------------------------------------------------

#define NN     100000
#define FEAT   128
#define NCLS   16
#define BN_EPS 1e-5f

typedef float v2f __attribute__((ext_vector_type(2)));
typedef float v8f __attribute__((ext_vector_type(8)));

// ---------------- utility: zero fill (ws is poisoned 0xAA) -----------------
__global__ void zero_kernel(float* __restrict__ p, size_t n) {
  size_t i = (size_t)blockIdx.x * blockDim.x + threadIdx.x;
  size_t stride = (size_t)gridDim.x * blockDim.x;
  for (; i < n; i += stride) p[i] = 0.0f;
}

// ---------------- degree: deg[dst] += 1 ------------------------------------
__global__ void degree_kernel(const int* __restrict__ dst,
                              float* __restrict__ deg, int nE) {
  int e = blockIdx.x * blockDim.x + threadIdx.x;
  if (e < nE) atomicAdd(&deg[dst[e]], 1.0f);
}

// ---------------- edge scatter-add: one wave32 per edge --------------------
// lane l moves features [4l, 4l+4): float4 gather (coalesced) + 4 f32 atomics
// into the destination row. Both h and agg are L2-resident (51.2MB each).
__global__ void __launch_bounds__(256)
scatter_kernel(const float* __restrict__ h, const int* __restrict__ src,
               const int* __restrict__ dst, float* __restrict__ agg, int nE) {
  int e = blockIdx.x * 8 + (threadIdx.x >> 5);
  if (e >= nE) return;
  int lane = threadIdx.x & 31;
  size_t s = (size_t)src[e] * FEAT;
  size_t d = (size_t)dst[e] * FEAT;
  float4 v = ((const float4*)(h + s))[lane];
  float* o = agg + d + (size_t)lane * 4;
  atomicAdd(o + 0, v.x);
  atomicAdd(o + 1, v.y);
  atomicAdd(o + 2, v.z);
  atomicAdd(o + 3, v.w);
}

// ---------------- fused SAGE layer: out = ReLU(BN(mean@Wl + b + h@Wr)) -----
// Block = 16 node rows; A tile [mean(16x128) | self(16x128)] staged in LDS
// (row stride 260 floats -> bank-conflict-free lane-strided reads).
// 8 waves x one 16x16 output tile each; K=256 via 64 v_wmma_f32_16x16x4_f32.
// In-place out==agg is safe: rows fully consumed into LDS before writeback.
#define SAGE_LDS_STRIDE 260
__global__ void __launch_bounds__(256)
sage_layer_kernel(const float* __restrict__ agg, const float* __restrict__ deg,
                  const float* __restrict__ hin,
                  const float* __restrict__ Wl, const float* __restrict__ Wr,
                  const float* __restrict__ bias,
                  const float* __restrict__ gma, const float* __restrict__ bta,
                  const float* __restrict__ rmu, const float* __restrict__ rvar,
                  float* __restrict__ out) {
  __shared__ float As[16 * SAGE_LDS_STRIDE];
  const int row0 = blockIdx.x * 16;
  const int tid  = threadIdx.x;

  // cooperative stage: 16 rows x 64 float4 (mean half | self half)
  for (int i = tid; i < 16 * 64; i += 256) {
    int r = i >> 6, c4 = i & 63;
    int row = row0 + r;
    float4 val;
    if (c4 < 32) {
      float4 a = ((const float4*)(agg + (size_t)row * FEAT))[c4];
      float inv = 1.0f / fmaxf(deg[row], 1.0f);
      val = float4{a.x * inv, a.y * inv, a.z * inv, a.w * inv};
    } else {
      val = ((const float4*)(hin + (size_t)row * FEAT))[c4 - 32];
    }
    *(float4*)(As + r * SAGE_LDS_STRIDE + c4 * 4) = val;
  }
  __syncthreads();

  const int lane = tid & 31;
  const int wv   = tid >> 5;         // 0..7 -> output column tile
  const int n0   = wv * 16;
  const int mm   = lane & 15;
  const int koff = (lane >> 4) << 1; // lanes 0-15: K+{0,1}; lanes 16-31: K+{2,3}
  const float* Am = As + mm * SAGE_LDS_STRIDE;

  v8f acc = {};
  // mean path (K 0..127) with Wl
  for (int k = 0; k < 128; k += 4) {
    v2f a, b;
    a.x = Am[k + koff];
    a.y = Am[k + koff + 1];
    b.x = Wl[(size_t)(k + koff) * FEAT + n0 + mm];
    b.y = Wl[(size_t)(k + koff + 1) * FEAT + n0 + mm];
    acc = __builtin_amdgcn_wmma_f32_16x16x4_f32(false, a, false, b,
                                                (short)0, acc, false, false);
  }
  // self path (K 128..255) with Wr
  for (int k = 0; k < 128; k += 4) {
    v2f a, b;
    a.x = Am[128 + k + koff];
    a.y = Am[128 + k + koff + 1];
    b.x = Wr[(size_t)(k + koff) * FEAT + n0 + mm];
    b.y = Wr[(size_t)(k + koff + 1) * FEAT + n0 + mm];
    acc = __builtin_amdgcn_wmma_f32_16x16x4_f32(false, a, false, b,
                                                (short)0, acc, false, false);
  }

  // epilogue: y = ReLU(((acc + bias) - mu) * g*rsqrt(var+eps) + beta)
  const int c = n0 + mm;
  const float scale = gma[c] * rsqrtf(rvar[c] + BN_EPS);
  const float shift = (bias[c] - rmu[c]) * scale + bta[c];
  const int rbase = (lane < 16) ? 0 : 8;   // C/D: vgpr r -> row r / r+8
  float* op = out + (size_t)row0 * FEAT + c;
#pragma unroll
  for (int i = 0; i < 8; ++i) {
    float y = acc[i] * scale + shift;
    op[(size_t)(rbase + i) * FEAT] = fmaxf(y, 0.0f);
  }
}

// ---------------- dense: out = (relu?)(h @ W + b), K=128, NOUT in {128,16} --
#define DENSE_LDS_STRIDE 132
template <int NOUT, bool RELU>
__global__ void dense_kernel(const float* __restrict__ hin,
                             const float* __restrict__ W,
                             const float* __restrict__ bias,
                             float* __restrict__ out) {
  __shared__ float As[16 * DENSE_LDS_STRIDE];
  const int row0 = blockIdx.x * 16;
  const int tid  = threadIdx.x;
  const int nthr = 32 * (NOUT / 16);

  for (int i = tid; i < 16 * 32; i += nthr) {
    int r = i >> 5, c4 = i & 31;
    *(float4*)(As + r * DENSE_LDS_STRIDE + c4 * 4) =
        ((const float4*)(hin + (size_t)(row0 + r) * FEAT))[c4];
  }
  __syncthreads();

  const int lane = tid & 31;
  const int wv   = tid >> 5;
  const int n0   = wv * 16;
  const int mm   = lane & 15;
  const int koff = (lane >> 4) << 1;
  const float* Am = As + mm * DENSE_LDS_STRIDE;

  v8f acc = {};
  for (int k = 0; k < 128; k += 4) {
    v2f a, b;
    a.x = Am[k + koff];
    a.y = Am[k + koff + 1];
    b.x = W[(size_t)(k + koff) * NOUT + n0 + mm];
    b.y = W[(size_t)(k + koff + 1) * NOUT + n0 + mm];
    acc = __builtin_amdgcn_wmma_f32_16x16x4_f32(false, a, false, b,
                                                (short)0, acc, false, false);
  }

  const int c = n0 + mm;
  const float bb = bias[c];
  const int rbase = (lane < 16) ? 0 : 8;
  float* op = out + (size_t)row0 * NOUT + c;
#pragma unroll
  for (int i = 0; i < 8; ++i) {
    float y = acc[i] + bb;
    if (RELU) y = fmaxf(y, 0.0f);
    op[(size_t)(rbase + i) * NOUT] = y;
  }
}

// ---------------------------------------------------------------------------
extern "C" void kernel_launch(void* const* d_in, const int* in_sizes, int n_in,
                              void* d_out, int out_size, void* d_ws,
                              size_t ws_size, hipStream_t stream) {
  (void)n_in; (void)out_size; (void)ws_size;

  const float* x   = (const float*)d_in[0];
  const int*   ei  = (const int*)d_in[1];
  const int    E_  = in_sizes[1] / 2;
  const int*   src = ei;
  const int*   dst = ei + E_;

  // per-layer params: W{i}l, b{i}, W{i}r, g{i}, be{i}, m{i}, v{i}
  const float* P[27];
  for (int i = 0; i < 27; ++i) P[i] = (const float*)d_in[i];
  const float* Wf = P[23];
  const float* bf = P[24];
  const float* Wc = P[25];
  const float* bc = P[26];

  // workspace: deg | bufA | bufB  (~103 MB)
  float* deg  = (float*)d_ws;
  float* bufA = deg + NN;
  float* bufB = bufA + (size_t)NN * FEAT;

  const size_t featN = (size_t)NN * FEAT;
  const int gemm_grid = NN / 16;           // 100000 % 16 == 0
  const int scat_grid = (E_ + 7) / 8;      // one wave32 per edge

  // degree (once per launch)
  zero_kernel<<<512, 256, 0, stream>>>(deg, (size_t)NN);
  degree_kernel<<<(E_ + 255) / 256, 256, 0, stream>>>(dst, deg, E_);

  // ---- Layer 1: agg=bufA (from x), out=bufB ----
  zero_kernel<<<4096, 256, 0, stream>>>(bufA, featN);
  scatter_kernel<<<scat_grid, 256, 0, stream>>>(x, src, dst, bufA, E_);
  sage_layer_kernel<<<gemm_grid, 256, 0, stream>>>(
      bufA, deg, x, P[2], P[4], P[3], P[5], P[6], P[7], P[8], bufB);

  // ---- Layer 2: agg=bufA (from bufB), out=bufA (in-place over agg) ----
  zero_kernel<<<4096, 256, 0, stream>>>(bufA, featN);
  scatter_kernel<<<scat_grid, 256, 0, stream>>>(bufB, src, dst, bufA, E_);
  sage_layer_kernel<<<gemm_grid, 256, 0, stream>>>(
      bufA, deg, bufB, P[9], P[11], P[10], P[12], P[13], P[14], P[15], bufA);

  // ---- Layer 3: agg=bufB (from bufA), out=bufB (in-place over agg) ----
  zero_kernel<<<4096, 256, 0, stream>>>(bufB, featN);
  scatter_kernel<<<scat_grid, 256, 0, stream>>>(bufA, src, dst, bufB, E_);
  sage_layer_kernel<<<gemm_grid, 256, 0, stream>>>(
      bufB, deg, bufA, P[16], P[18], P[17], P[19], P[20], P[21], P[22], bufB);

  // ---- head: ReLU(h3 @ Wf + bf) -> bufA ; bufA @ Wc + bc -> d_out ----
  dense_kernel<128, true><<<gemm_grid, 256, 0, stream>>>(bufB, Wf, bf, bufA);
  dense_kernel<16, false><<<gemm_grid, 32, 0, stream>>>(bufA, Wc, bc,
                                                        (float*)d_out);
}